// PIGR_20332375179634
// MI455X (gfx1250) — compile-verified
//
#include <hip/hip_runtime.h>
#include <math.h>

// ---------------------------------------------------------------------------
// PIGR session model for MI455X (gfx1250, wave32).
// All GEMMs use V_WMMA_F32_16X16X4_F32 (fp32 A/B/C) — exact fp32 math on the
// matrix pipe. Workload is HBM/L2-bandwidth bound (~400 MB moved, ~19 GFLOP),
// so fp32 WMMA is the right precision/throughput point.
// ---------------------------------------------------------------------------

#define BB   256
#define LL   200
#define LP   208          // L padded to 13 * 16 tiles
#define HH   128
#define VV   50000
#define TOPK 20
#define NEGF (-1e10f)

typedef float v2f __attribute__((ext_vector_type(2)));
typedef float v8f __attribute__((ext_vector_type(8)));

// D = A(16x4) * B(4x16) + C, fp32.  Layout (ISA 7.12.2):
//   A: lane<16 -> row=lane, v0=K0,v1=K1 ; lane>=16 -> row=lane-16, v0=K2,v1=K3
//   B: symmetric (col=lane&15, K per half)
//   C/D: vgpr r -> M = r + 8*(lane>>4), N = lane&15
__device__ inline v8f wmma4(v2f a, v2f b, v8f c) {
  return __builtin_amdgcn_wmma_f32_16x16x4_f32(false, a, false, b, (short)0, c,
                                               false, false);
}

__device__ inline float softplus1(float x) {       // torch Softplus(threshold=1)
  return x > 1.0f ? x : log1pf(expf(fminf(x, 1.0f)));
}

__device__ inline float waveRedSum(float v) {
  #pragma unroll
  for (int o = 16; o > 0; o >>= 1) v += __shfl_down(v, o, 32);
  return v;
}
__device__ inline float waveRedMax(float v) {
  #pragma unroll
  for (int o = 16; o > 0; o >>= 1) v = fmaxf(v, __shfl_down(v, o, 32));
  return v;
}
// Every thread returns the same value; block dim must be a multiple of 32.
__device__ inline float blockRedSum(float v, float* sh) {
  int tid = threadIdx.x, nw = blockDim.x >> 5;
  v = waveRedSum(v);
  __syncthreads();
  if ((tid & 31) == 0) sh[tid >> 5] = v;
  __syncthreads();
  float r = 0.f;
  for (int i = 0; i < nw; ++i) r += sh[i];
  __syncthreads();
  return r;
}
__device__ inline float blockRedMax(float v, float* sh) {
  int tid = threadIdx.x, nw = blockDim.x >> 5;
  v = waveRedMax(v);
  __syncthreads();
  if ((tid & 31) == 0) sh[tid >> 5] = v;
  __syncthreads();
  float r = -3.4e38f;
  for (int i = 0; i < nw; ++i) r = fmaxf(r, sh[i]);
  __syncthreads();
  return r;
}

// ---- K1: gather hidden = emb_w[seq] into padded [B,LP,H]; row L2 norms -----
__global__ __launch_bounds__(HH) void k_gather(const int* __restrict__ seq,
                                               const float* __restrict__ emb,
                                               float* __restrict__ hidden,
                                               float* __restrict__ nrm) {
  __shared__ float sh[8];
  int l = blockIdx.x, b = blockIdx.y, h = threadIdx.x;
  float v = 0.f;
  if (l < LL) {
    int t = seq[b * LL + l];
    v = emb[(size_t)t * HH + h];
  }
  hidden[((size_t)b * LP + l) * HH + h] = v;
  float ss = blockRedSum(v * v, sh);
  if (h == 0) nrm[b * LP + l] = sqrtf(ss);
}

// ---- K2: slen / e_s / zero_layer (layer_collect[0]) ------------------------
__global__ __launch_bounds__(HH) void k_stats(const float* __restrict__ hidden,
                                              const int* __restrict__ rev,
                                              float* __restrict__ e_s,
                                              float* __restrict__ lc0) {
  int b = blockIdx.x, h = threadIdx.x;
  int sl = rev[b * LL];  // reverse_lengths[b,0] == len (valid prefix length)
  float acc = 0.f;
  for (int l = 0; l < LL; ++l) acc += hidden[((size_t)b * LP + l) * HH + h];
  lc0[b * HH + h] = acc / (float)sl;                 // pads are emb row 0 == 0
  e_s[b * HH + h] = hidden[((size_t)b * LP + (sl - 1)) * HH + h];
}

// ---- K3: base sim (cos + diag) and softplus noise, fused dual WMMA ---------
__global__ __launch_bounds__(32) void k_sim(const float* __restrict__ hidden,
                                            const float* __restrict__ wnoise,
                                            const int* __restrict__ rev,
                                            const float* __restrict__ nrm,
                                            const float* __restrict__ clw,
                                            float* __restrict__ base,
                                            float* __restrict__ noise) {
  int tm = blockIdx.x, tn = blockIdx.y, b = blockIdx.z;
  int lane = threadIdx.x, half = lane >> 4, idx = lane & 15;
  const float* Hb = hidden + (size_t)b * LP * HH;
  int arow = tm * 16 + idx;                 // A row (sim row l)
  int bcol = tn * 16 + idx;                 // B col (sim col m)
  int ridx = (bcol < LL) ? rev[b * LL + bcol] : 0;
  const float* Wrow = wnoise + (size_t)ridx * HH;
  v8f cs = {0, 0, 0, 0, 0, 0, 0, 0};
  v8f cn = {0, 0, 0, 0, 0, 0, 0, 0};
  for (int k = 0; k < HH; k += 4) {
    int kk = k + 2 * half;
    v2f a, bh, bw;
    a[0]  = Hb[(size_t)arow * HH + kk];     a[1]  = Hb[(size_t)arow * HH + kk + 1];
    bh[0] = Hb[(size_t)bcol * HH + kk];     bh[1] = Hb[(size_t)bcol * HH + kk + 1];
    bw[0] = Wrow[kk];                       bw[1] = Wrow[kk + 1];
    cs = wmma4(a, bh, cs);                  // hidden . hidden^T
    cn = wmma4(a, bw, cn);                  // hidden . wn^T
  }
  float nn = nrm[b * LP + tn * 16 + idx];   // column norm for this lane
  #pragma unroll
  for (int r = 0; r < 8; ++r) {
    int m = tm * 16 + r + 8 * half;         // output row
    int n = tn * 16 + idx;                  // output col
    float s = cs[r] / (nrm[b * LP + m] * nn + 1e-6f);
    if (m == n && m < LL) s += clw[rev[b * LL + m]];   // diag_embed term
    size_t o = ((size_t)b * LP + m) * LP + n;
    base[o] = s;
    noise[o] = softplus1(cn[r]);
  }
}

// ---- K4a: per-row top-K selection -> edge bytes (edge[b][row][col]) --------
__global__ __launch_bounds__(256) void k_topk(const float* __restrict__ base,
                                              const float* __restrict__ noise,
                                              const float* __restrict__ coffL,
                                              const int* __restrict__ rev,
                                              unsigned char* __restrict__ edge) {
  __shared__ float svals[LP];
  __shared__ float sv[256];
  __shared__ int   si[256];
  int r = blockIdx.x, b = blockIdx.y, tid = threadIdx.x;
  int sl = rev[b * LL];
  unsigned char* erow = edge + ((size_t)b * LP + r) * LP;
  if (tid < LP) erow[tid] = 0;
  if (r >= sl) return;                      // all-NEG row -> no edges (uniform)
  if (tid < LP) {
    float v = NEGF;
    if (tid < sl)
      v = base[((size_t)b * LP + r) * LP + tid] +
          coffL[((size_t)b * LL + r) * LL + tid] *
              noise[((size_t)b * LP + r) * LP + tid];
    svals[tid] = v;
  }
  __syncthreads();
  for (int it = 0; it < TOPK; ++it) {
    float mv = (tid < LP) ? svals[tid] : -3.4e38f;
    sv[tid] = mv; si[tid] = tid;
    __syncthreads();
    #pragma unroll
    for (int s = 128; s > 0; s >>= 1) {
      if (tid < s) {
        float v2 = sv[tid + s]; int i2 = si[tid + s];
        if (v2 > sv[tid] || (v2 == sv[tid] && i2 < si[tid])) { sv[tid] = v2; si[tid] = i2; }
      }
      __syncthreads();
    }
    int best = si[0]; float bv = sv[0];
    __syncthreads();
    if (tid == 0) {
      if (bv > NEGF) erow[best] = 1;        // edges = sel * (msim > NEG)
      svals[best] = -3.4e38f;
    }
    __syncthreads();
  }
}

// ---- K4b: ssm[b,l,m] = edge[b,m,l] * softmax_row_l(msim)[m]; cmask row sum -
__global__ __launch_bounds__(256) void k_ssm(const float* __restrict__ base,
                                             const float* __restrict__ noise,
                                             const float* __restrict__ coffL,
                                             const int* __restrict__ rev,
                                             const unsigned char* __restrict__ edge,
                                             float* __restrict__ ssm,
                                             float* __restrict__ cmask) {
  __shared__ float sh[8];
  int l = blockIdx.x, b = blockIdx.y, tid = threadIdx.x;
  int sl = rev[b * LL];
  float* srow = ssm + ((size_t)b * LP + l) * LP;
  if (l >= sl) {                            // uniform branch
    if (tid < LP) srow[tid] = 0.f;
    if (tid == 0) cmask[b * LP + l] = 0.f;
    return;
  }
  float v = NEGF;
  if (tid < sl)
    v = base[((size_t)b * LP + l) * LP + tid] +
        coffL[((size_t)b * LL + l) * LL + tid] *
            noise[((size_t)b * LP + l) * LP + tid];
  float mx = blockRedMax(v, sh);
  float e = expf(v - mx);                   // NEG entries underflow to 0 exactly
  float s = blockRedSum(e, sh);
  float p = e / s;
  float ed = (tid < LP) ? (float)edge[((size_t)b * LP + tid) * LP + l] : 0.f;
  float sv = p * ed;
  if (tid < LP) srow[tid] = sv;
  float cm = blockRedSum(sv, sh);
  if (tid == 0) cmask[b * LP + l] = cm;
}

// ---- K4c: cmask /= max(||cmask||_2, 1e-12) ---------------------------------
__global__ __launch_bounds__(256) void k_cmask_norm(float* __restrict__ cmask) {
  __shared__ float sh[8];
  int b = blockIdx.x, tid = threadIdx.x;
  float v = (tid < LP) ? cmask[b * LP + tid] : 0.f;
  float ss = blockRedSum(v * v, sh);
  float d = fmaxf(sqrtf(ss), 1e-12f);
  if (tid < LP) cmask[b * LP + tid] = v / d;
}

// ---- K4d: update = ssm @ hidden  ([LP x LP] x [LP x H]), WMMA --------------
__global__ __launch_bounds__(32) void k_update(const float* __restrict__ ssm,
                                               const float* __restrict__ hidden,
                                               float* __restrict__ update) {
  int tn = blockIdx.x, tm = blockIdx.y, b = blockIdx.z;
  int lane = threadIdx.x, half = lane >> 4, idx = lane & 15;
  const float* Sb = ssm + (size_t)b * LP * LP;
  const float* Hb = hidden + (size_t)b * LP * HH;
  int arow = tm * 16 + idx;
  int ncol = tn * 16 + idx;
  v8f c = {0, 0, 0, 0, 0, 0, 0, 0};
  for (int k = 0; k < LP; k += 4) {
    int kk = k + 2 * half;
    v2f a, bb;
    a[0]  = Sb[(size_t)arow * LP + kk]; a[1] = Sb[(size_t)arow * LP + kk + 1];
    bb[0] = Hb[(size_t)kk * HH + ncol]; bb[1] = Hb[(size_t)(kk + 1) * HH + ncol];
    c = wmma4(a, bb, c);
  }
  #pragma unroll
  for (int r = 0; r < 8; ++r)
    update[((size_t)b * LP + tm * 16 + r + 8 * half) * HH + ncol] = c[r];
}

// ---- K4e: pre = relu(u@W1+b1)@W2 + b2 + u  (two WMMA stages through LDS) ---
// (the reference's (slen>1) gate for layer>0 is always 1 since len >= 50)
__global__ __launch_bounds__(256) void k_ffn(const float* __restrict__ update,
                                             const float* __restrict__ w1,
                                             const float* __restrict__ b1,
                                             const float* __restrict__ w2,
                                             const float* __restrict__ b2,
                                             float* __restrict__ pre) {
  __shared__ float sfeed[16][HH];           // 8 KB
  int tt = blockIdx.x, b = blockIdx.y;
  int tid = threadIdx.x, w = tid >> 5, lane = tid & 31;
  int half = lane >> 4, idx = lane & 15;
  const float* U = update + ((size_t)b * LP + tt * 16) * HH;
  int n = w * 16 + idx;                     // this wave's output column
  v8f c1 = {0, 0, 0, 0, 0, 0, 0, 0};
  for (int k = 0; k < HH; k += 4) {
    int kk = k + 2 * half;
    v2f a, bb;
    a[0]  = U[(size_t)idx * HH + kk]; a[1] = U[(size_t)idx * HH + kk + 1];
    bb[0] = w1[(size_t)kk * HH + n];  bb[1] = w1[(size_t)(kk + 1) * HH + n];
    c1 = wmma4(a, bb, c1);
  }
  float bias1 = b1[n];
  #pragma unroll
  for (int r = 0; r < 8; ++r)
    sfeed[r + 8 * half][n] = fmaxf(c1[r] + bias1, 0.f);
  __syncthreads();
  v8f c2 = {0, 0, 0, 0, 0, 0, 0, 0};
  for (int k = 0; k < HH; k += 4) {
    int kk = k + 2 * half;
    v2f a, bb;
    a[0]  = sfeed[idx][kk];          a[1] = sfeed[idx][kk + 1];
    bb[0] = w2[(size_t)kk * HH + n]; bb[1] = w2[(size_t)(kk + 1) * HH + n];
    c2 = wmma4(a, bb, c2);
  }
  float bias2 = b2[n];
  #pragma unroll
  for (int r = 0; r < 8; ++r) {
    int m = r + 8 * half;
    pre[((size_t)b * LP + tt * 16 + m) * HH + n] =
        c2[r] + bias2 + U[(size_t)m * HH + n];       // residual
  }
}

// ---- K4f: LayerNorm + cmask-weighted sum over tokens -> lc[layer+1] --------
__global__ __launch_bounds__(HH) void k_ln_accum(const float* __restrict__ pre,
                                                 const float* __restrict__ cmask,
                                                 const float* __restrict__ g,
                                                 const float* __restrict__ bta,
                                                 float* __restrict__ lc_out) {
  __shared__ float sh[8];
  int b = blockIdx.x, h = threadIdx.x;
  float acc = 0.f, gg = g[h], bb = bta[h];
  for (int l = 0; l < LL; ++l) {
    float cm = cmask[b * LP + l];           // uniform across block
    if (cm == 0.f) continue;
    float p = pre[((size_t)b * LP + l) * HH + h];
    float mu = blockRedSum(p, sh) * (1.0f / HH);
    float d = p - mu;
    float var = blockRedSum(d * d, sh) * (1.0f / HH);
    acc += (d * rsqrtf(var + 1e-5f) * gg + bb) * cm;
  }
  lc_out[b * HH + h] = acc;
}

// ---- K5: e_c mean, gate, sess ----------------------------------------------
__global__ __launch_bounds__(HH) void k_gate(const float* __restrict__ lc,
                                             const float* __restrict__ e_s,
                                             const float* __restrict__ gw,
                                             float* __restrict__ sess) {
  __shared__ float ec[HH], es[HH];
  int b = blockIdx.x, h = threadIdx.x;
  ec[h] = (lc[b * HH + h] + lc[(size_t)BB * HH + b * HH + h] +
           lc[(size_t)2 * BB * HH + b * HH + h]) * (1.0f / 3.0f);
  es[h] = e_s[b * HH + h];
  __syncthreads();
  float a = 0.f;
  for (int j = 0; j < HH; ++j) a += ec[j] * gw[(size_t)j * HH + h];
  for (int j = 0; j < HH; ++j) a += es[j] * gw[(size_t)(HH + j) * HH + h];
  a = 1.0f / (1.0f + expf(-a));
  sess[b * HH + h] = a * ec[h] + (1.0f - a) * es[h];
}

// ---- K6: scores = sess @ emb_w^T  ([256x128] x [128x50000]), WMMA ----------
__global__ __launch_bounds__(32) void k_scores(const float* __restrict__ sess,
                                               const float* __restrict__ emb,
                                               float* __restrict__ out) {
  int tv = blockIdx.x, tb = blockIdx.y;
  int lane = threadIdx.x, half = lane >> 4, idx = lane & 15;
  int brow = tb * 16 + idx;
  int vcol = tv * 16 + idx;
  v8f c = {0, 0, 0, 0, 0, 0, 0, 0};
  for (int k = 0; k < HH; k += 4) {
    int kk = k + 2 * half;
    v2f a, bb;
    a[0]  = sess[(size_t)brow * HH + kk]; a[1] = sess[(size_t)brow * HH + kk + 1];
    bb[0] = emb[(size_t)vcol * HH + kk];  bb[1] = emb[(size_t)vcol * HH + kk + 1];
    c = wmma4(a, bb, c);
  }
  #pragma unroll
  for (int r = 0; r < 8; ++r)
    out[(size_t)(tb * 16 + r + 8 * half) * VV + tv * 16 + idx] = c[r];
}

__global__ void k_scalar(float* out) { out[(size_t)BB * VV] = 0.0f; }

// ---------------------------------------------------------------------------
extern "C" void kernel_launch(void* const* d_in, const int* in_sizes, int n_in,
                              void* d_out, int out_size, void* d_ws, size_t ws_size,
                              hipStream_t stream) {
  (void)in_sizes; (void)n_in; (void)out_size; (void)ws_size;
  const int*   seq    = (const int*)d_in[0];
  const int*   rev    = (const int*)d_in[2];   // reverse_lengths
  const float* emb    = (const float*)d_in[3];
  const float* wnoise = (const float*)d_in[4];
  const float* clw    = (const float*)d_in[5];
  const float* w1     = (const float*)d_in[6];
  const float* b1     = (const float*)d_in[7];
  const float* w2     = (const float*)d_in[8];
  const float* b2     = (const float*)d_in[9];
  const float* ln_g   = (const float*)d_in[10];
  const float* ln_b   = (const float*)d_in[11];
  const float* gw     = (const float*)d_in[12];
  const float* coff   = (const float*)d_in[13];
  float* out = (float*)d_out;

  char* w = (char*)d_ws;
  size_t off = 0;
  auto carve = [&](size_t bytes) {
    void* p = w + off;
    off = (off + bytes + 255) & ~(size_t)255;
    return p;
  };
  float* hidden = (float*)carve((size_t)BB * LP * HH * 4);   // 27.3 MB
  float* base   = (float*)carve((size_t)BB * LP * LP * 4);   // 44.3 MB
  float* noise  = (float*)carve((size_t)BB * LP * LP * 4);   // 44.3 MB
  float* ssm    = (float*)carve((size_t)BB * LP * LP * 4);   // 44.3 MB
  float* update = (float*)carve((size_t)BB * LP * HH * 4);   // 27.3 MB
  float* pre    = (float*)carve((size_t)BB * LP * HH * 4);   // 27.3 MB
  unsigned char* edge = (unsigned char*)carve((size_t)BB * LP * LP); // 11.1 MB
  float* nrm    = (float*)carve((size_t)BB * LP * 4);
  float* e_s    = (float*)carve((size_t)BB * HH * 4);
  float* lc     = (float*)carve((size_t)3 * BB * HH * 4);
  float* cmask  = (float*)carve((size_t)BB * LP * 4);
  float* sess   = (float*)carve((size_t)BB * HH * 4);

  k_gather<<<dim3(LP, BB), HH, 0, stream>>>(seq, emb, hidden, nrm);
  k_stats <<<BB, HH, 0, stream>>>(hidden, rev, e_s, lc);
  k_sim   <<<dim3(LP / 16, LP / 16, BB), 32, 0, stream>>>(hidden, wnoise, rev,
                                                          nrm, clw, base, noise);
  for (int layer = 0; layer < 2; ++layer) {
    const float* coffL = coff + (size_t)layer * BB * LL * LL;
    k_topk      <<<dim3(LP, BB), 256, 0, stream>>>(base, noise, coffL, rev, edge);
    k_ssm       <<<dim3(LP, BB), 256, 0, stream>>>(base, noise, coffL, rev, edge,
                                                   ssm, cmask);
    k_cmask_norm<<<BB, 256, 0, stream>>>(cmask);
    k_update    <<<dim3(HH / 16, LP / 16, BB), 32, 0, stream>>>(ssm, hidden, update);
    k_ffn       <<<dim3(LP / 16, BB), 256, 0, stream>>>(update, w1, b1, w2, b2, pre);
    k_ln_accum  <<<BB, HH, 0, stream>>>(pre, cmask, ln_g, ln_b,
                                        lc + (size_t)(layer + 1) * BB * HH);
  }
  k_gate  <<<BB, HH, 0, stream>>>(lc, e_s, gw, sess);
  k_scores<<<dim3(VV / 16, BB / 16), 32, 0, stream>>>(sess, emb, out);
  k_scalar<<<1, 1, 0, stream>>>(out);
}